// BidirectionalReadout_60181081752052
// MI455X (gfx1250) — compile-verified
//
#include <hip/hip_runtime.h>
#include <hip/hip_bf16.h>
#include <math.h>

#define B_   2
#define L_   2048
#define D_   256
#define N_   64
#define ROWS (B_ * L_)   // 4096
#define KD   D_          // GEMM reduction dim = 256

typedef float v2f __attribute__((ext_vector_type(2)));
typedef float v8f __attribute__((ext_vector_type(8)));

__device__ __forceinline__ float softplus_f(float v) {
    // stable, branch-free: max(v,0) + log1p(exp(-|v|)); exp arg <= 0 so __expf is safe
    return fmaxf(v, 0.0f) + log1pf(__expf(-fabsf(v)));
}

// ---------------------------------------------------------------------------
// Projection GEMM via V_WMMA_F32_16X16X4_F32, templated on output kind so the
// weight leading dimension is a compile-time constant (immediate-offset loads
// instead of runtime 64-bit address math in the K-loop).
//   DELTA=true : LD=256, cols = f_Wdt | b_Wdt; epilogue fuses bias+softplus and
//                packs float2{delta, delta*x} for the scan.
//   DELTA=false: LD=64,  cols = f_WB | b_WB; raw store.
// One wave per 16x16 tile; 64 WMMA steps over K=256.
// ---------------------------------------------------------------------------
template <bool DELTA>
__global__ void proj_kernel(const float* __restrict__ x,
                            const float* __restrict__ Wf, const float* __restrict__ biasf,
                            const float* __restrict__ Wb, const float* __restrict__ biasb,
                            float* __restrict__ dstf, float* __restrict__ dstb)
{
    constexpr int LD  = DELTA ? D_ : N_;   // weight leading dim (compile-time)
    constexpr int TPD = LD / 16;           // 16-col tiles per direction

    const int lane    = threadIdx.x;       // 0..31 (wave32)
    const int rowTile = blockIdx.x;        // 0..255
    int ct            = blockIdx.y;        // 0..2*TPD-1
    const bool back   = ct >= TPD;
    if (back) ct -= TPD;

    const float* W    = back ? Wb   : Wf;
    const float* bias = back ? biasb : biasf;
    float*       dst  = back ? dstb : dstf;
    const int colOff  = ct * 16;

    const int m  = lane & 15;              // A row / B col within tile
    const int hi = lane >> 4;              // K-pair select (0/1 vs 2/3)
    const int rowBase = rowTile * 16;

    const float* aRow = x + (size_t)(rowBase + m) * KD;
    const float* bCol = W + colOff + m;

    __builtin_prefetch(bCol, 0, 1);        // global_prefetch_b8

    v8f acc = {};
#pragma unroll 4
    for (int k0 = 0; k0 < KD; k0 += 4) {
        const int ka = k0 + hi * 2;
        v2f a = *(const v2f*)(aRow + ka);  // contiguous b64 load of x row
        v2f b;
        b.x = bCol[ka * LD];               // constant-stride: immediate offsets
        b.y = bCol[(ka + 1) * LD];
        acc = __builtin_amdgcn_wmma_f32_16x16x4_f32(
            /*neg_a=*/false, a, /*neg_b=*/false, b,
            /*c_mod=*/(short)0, acc, /*reuse_a=*/false, /*reuse_b=*/false);
    }

    // C/D layout: VGPR r -> M = r + 8*hi ; N = lane&15
    const int col = colOff + m;
    if constexpr (DELTA) {
        const float bv = bias[col];
#pragma unroll
        for (int r = 0; r < 8; ++r) {
            const int   row = rowBase + r + hi * 8;
            const float dlt = softplus_f(acc[r] + bv);
            const float xv  = x[(size_t)row * D_ + col];
            v2f du; du.x = dlt; du.y = dlt * xv;     // {delta, delta*x}
            *(v2f*)(dst + 2 * ((size_t)row * D_ + col)) = du;
        }
    } else {
#pragma unroll
        for (int r = 0; r < 8; ++r) {
            const int row = rowBase + r + hi * 8;
            dst[(size_t)row * N_ + col] = acc[r];
        }
    }
}

// ---------------------------------------------------------------------------
// The only two C rows ever needed per direction.
// blockIdx.x = dir*2 + b ; thread n computes x[b, tsel, :] . WC[:, n]
// ---------------------------------------------------------------------------
__global__ void cmlast_kernel(const float* __restrict__ x,
                              const float* __restrict__ fWC, const float* __restrict__ bWC,
                              float* __restrict__ CmLast)
{
    const int n   = threadIdx.x;      // 0..63
    const int blk = blockIdx.x;       // 0..3
    const int dir = blk >> 1, b = blk & 1;
    const float* WC = dir ? bWC : fWC;
    const int tsel  = dir ? 0 : (L_ - 1);
    const float* xr = x + ((size_t)b * L_ + tsel) * D_;
    float acc = 0.0f;
    for (int k = 0; k < D_; ++k) acc += xr[k] * WC[k * N_ + n];
    CmLast[blk * N_ + n] = acc;
}

// ---------------------------------------------------------------------------
// Selective scan, final state only. Grid (D, B, dir), 64 threads (one state n
// each). Per step: one block-uniform b64 load of {delta, delta*x} (scalarizable
// by uniformity analysis) + one coalesced b32 of Bm, then mul -> v_exp -> fma.
// Backward direction walks t in reverse. LDS reduction over n, D-skip at the
// endpoint.
// ---------------------------------------------------------------------------
__global__ void scan_kernel(const float* __restrict__ x,
                            const float* __restrict__ fAlog, const float* __restrict__ bAlog,
                            const float* __restrict__ fDp,   const float* __restrict__ bDp,
                            const float* __restrict__ du_f,  const float* __restrict__ du_b,
                            const float* __restrict__ Bm_f,  const float* __restrict__ Bm_b,
                            const float* __restrict__ CmLast,
                            float* __restrict__ ys)
{
    __shared__ float red[64];
    const int n   = threadIdx.x;   // 0..63
    const int d   = blockIdx.x;    // 0..255
    const int b   = blockIdx.y;    // 0..1
    const int dir = blockIdx.z;    // 0=fwd, 1=bwd

    const float*  Alog = dir ? bAlog : fAlog;
    const float2* duP  = (const float2*)(dir ? du_b : du_f);
    const float*  Bm   = dir ? Bm_b : Bm_f;
    const float   A    = -__expf(Alog[d * N_ + n]);
    const size_t  base = (size_t)b * L_;

    float h = 0.0f;
    if (dir == 0) {
        for (int t = 0; t < L_; ++t) {
            const size_t rt  = base + t;
            const float2 du2 = duP[rt * D_ + d];   // {delta, delta*x}, uniform
            const float  bm  = Bm[rt * N_ + n];
            h = __expf(du2.x * A) * h + du2.y * bm;
        }
    } else {
        for (int t = L_ - 1; t >= 0; --t) {        // reversed sequence
            const size_t rt  = base + t;
            const float2 du2 = duP[rt * D_ + d];
            const float  bm  = Bm[rt * N_ + n];
            h = __expf(du2.x * A) * h + du2.y * bm;
        }
    }

    red[n] = h * CmLast[(dir * 2 + b) * N_ + n];
    __syncthreads();
    for (int s = 32; s > 0; s >>= 1) {
        if (n < s) red[n] += red[n + s];
        __syncthreads();
    }
    if (n == 0) {
        const int    tl = dir ? 0 : (L_ - 1);
        const float* Dp = dir ? bDp : fDp;
        ys[(dir * 2 + b) * D_ + d] = red[0] + Dp[d] * x[(base + tl) * D_ + d];
    }
}

// ---------------------------------------------------------------------------
// Gated MLP head + output projection (exact-erf GELU). One block per batch
// row, 256 threads (one per channel), LDS-staged dots.
// ---------------------------------------------------------------------------
__global__ void head_kernel(const float* __restrict__ ys,
                            const float* __restrict__ W1, const float* __restrict__ b1,
                            const float* __restrict__ W2, const float* __restrict__ b2,
                            const float* __restrict__ W3, const float* __restrict__ b3,
                            const float* __restrict__ Wout, const float* __restrict__ bout,
                            float* __restrict__ out)
{
    __shared__ float xc[D_];
    __shared__ float sv[D_];
    const int d = threadIdx.x;
    const int b = blockIdx.x;

    const float v = 0.5f * (ys[(0 * 2 + b) * D_ + d] + ys[(1 * 2 + b) * D_ + d]);
    xc[d] = v;
    __syncthreads();

    const float* Ws[3] = {W1, W2, W3};
    const float* bs[3] = {b1, b2, b3};
    float s = 0.0f;
    for (int i = 0; i < 3; ++i) {
        float acc = bs[i][d];
        for (int k = 0; k < D_; ++k) acc += xc[k] * Ws[i][k * D_ + d];
        const float g = 0.5f * acc * (1.0f + erff(acc * 0.70710678118654752f));
        s += g * v;
    }
    sv[d] = s;
    __syncthreads();

    float acc = bout[d];
    for (int k = 0; k < D_; ++k) acc += sv[k] * Wout[k * D_ + d];
    out[b * D_ + d] = acc;
}

// ---------------------------------------------------------------------------
extern "C" void kernel_launch(void* const* d_in, const int* in_sizes, int n_in,
                              void* d_out, int out_size, void* d_ws, size_t ws_size,
                              hipStream_t stream)
{
    (void)in_sizes; (void)n_in; (void)out_size; (void)ws_size;
    const float* x     = (const float*)d_in[0];
    const float* fAlog = (const float*)d_in[1];
    const float* fDp   = (const float*)d_in[2];
    const float* fWdt  = (const float*)d_in[3];
    const float* fbdt  = (const float*)d_in[4];
    const float* fWB   = (const float*)d_in[5];
    const float* fWC   = (const float*)d_in[6];
    const float* bAlog = (const float*)d_in[7];
    const float* bDp   = (const float*)d_in[8];
    const float* bWdt  = (const float*)d_in[9];
    const float* bbdt  = (const float*)d_in[10];
    const float* bWB   = (const float*)d_in[11];
    const float* bWC   = (const float*)d_in[12];
    const float* W1    = (const float*)d_in[13];
    const float* W2    = (const float*)d_in[14];
    const float* W3    = (const float*)d_in[15];
    const float* Wout  = (const float*)d_in[16];
    const float* b1    = (const float*)d_in[17];
    const float* b2    = (const float*)d_in[18];
    const float* b3    = (const float*)d_in[19];
    const float* bout  = (const float*)d_in[20];

    // Scratch layout (floats), total ~19 MB (L2 is 192 MB)
    float* ws   = (float*)d_ws;
    float* du_f = ws;  ws += (size_t)2 * ROWS * D_;   // float2 {delta, delta*x}
    float* du_b = ws;  ws += (size_t)2 * ROWS * D_;
    float* Bm_f = ws;  ws += (size_t)ROWS * N_;
    float* Bm_b = ws;  ws += (size_t)ROWS * N_;
    float* CmLast = ws; ws += 4 * N_;
    float* ysb  = ws;  ws += 4 * D_;

    proj_kernel<true><<<dim3(ROWS / 16, 32), 32, 0, stream>>>(
        x, fWdt, fbdt, bWdt, bbdt, du_f, du_b);
    proj_kernel<false><<<dim3(ROWS / 16, 8), 32, 0, stream>>>(
        x, fWB, nullptr, bWB, nullptr, Bm_f, Bm_b);
    cmlast_kernel<<<4, 64, 0, stream>>>(x, fWC, bWC, CmLast);
    scan_kernel<<<dim3(D_, B_, 2), 64, 0, stream>>>(
        x, fAlog, bAlog, fDp, bDp, du_f, du_b, Bm_f, Bm_b, CmLast, ysb);
    head_kernel<<<B_, D_, 0, stream>>>(
        ysb, W1, b1, W2, b2, W3, b3, Wout, bout, (float*)d_out);
}